// MARN1_onlysp_76046690943294
// MI455X (gfx1250) — compile-verified
//
#include <hip/hip_runtime.h>
#include <hip/hip_bf16.h>
#include <math.h>

// ---------------------------------------------------------------------------
// MARN inference for MI455X (gfx1250, wave32).
// GEMMs use V_WMMA_F32_16X16X4_F32 (full fp32 matrix pipe), one wave per
// 16x32 output tile (two accumulators -> independent back-to-back WMMAs).
// Recurrence runs as a persistent kernel: grid(32 samples, 2 directions).
// ---------------------------------------------------------------------------

typedef __attribute__((ext_vector_type(2))) float v2f;
typedef __attribute__((ext_vector_type(8))) float v8f;

#define TT   512
#define NN   32
#define DR   1024
#define DL   100
#define DA   100
#define DHH  128
#define NHD  8
#define DKH  40
#define PP   9
#define ROWS (TT*NN)

__device__ inline float wred_add(float v){
#pragma unroll
  for (int o=16;o>0;o>>=1) v += __shfl_xor(v,o,32);
  return v;
}
__device__ inline float wred_max(float v){
#pragma unroll
  for (int o=16;o>0;o>>=1) v = fmaxf(v,__shfl_xor(v,o,32));
  return v;
}
__device__ inline float sigf(float x){ return 1.0f/(1.0f+__expf(-x)); }

// ---------------------------------------------------------------------------
// WMMA GEMM: C[M,Ncols] = act(alpha*A[M,K] @ B[K,Ncols] + bias)
//   MODE 0: A row-major (lda stride, +a_col_off)
//   MODE 1: logical row r=(n*512+t) maps to physical row (t*32+n)  (TN layout)
// K is a multiple of 4 everywhere. Per lane, the A fragment is k-contiguous
// -> single aligned 8B load. B edge tile handled by address clamp (garbage
// only lands in never-stored lanes/columns).
// ---------------------------------------------------------------------------
struct GemmArgs {
  const float* A; long lda; int a_mode; long a_col_off;
  const float* B; const float* bias; const float* alpha;
  float* C; long ldc;
  int M, Ncols, K, relu;
  const float* f0; const float* f1; const float* f2; const float* f3;
};

template<int MODE>
__global__ __launch_bounds__(32) void k_gemm_wmma(GemmArgs g){
  const int lane = threadIdx.x;
  const int m16  = lane & 15;
  const int kb   = (lane >> 4) << 1;        // lanes 0-15: K=0,1 ; lanes 16-31: K=2,3
  const int row  = blockIdx.x*16 + m16;     // A-fragment row (logical)
  const int col0 = blockIdx.y*32 + m16;
  const int col1 = col0 + 16;
  const bool c0 = col0 < g.Ncols, c1 = col1 < g.Ncols;
  const float alpha = g.alpha ? g.alpha[0] : 1.0f;

  const float* Ap;
  if (MODE==0){ Ap = g.A + (long)row*g.lda + g.a_col_off; }
  else        { int n=row>>9, t=row&511; Ap = g.A + ((long)t*NN+n)*g.lda + g.a_col_off; }
  const long ldb = g.Ncols;
  const float* B0 = g.B + (c0 ? col0 : (g.Ncols-1));
  const float* B1 = g.B + (c1 ? col1 : (g.Ncols-1));

  v8f acc0 = {}, acc1 = {};
#pragma unroll 4
  for (int k0=0; k0<g.K; k0+=4){
    const int k = k0 + kb;
    v2f a = *(const v2f*)(Ap + k);
    a.x *= alpha; a.y *= alpha;
    v2f b0, b1;
    b0.x = B0[(long)k*ldb];  b0.y = B0[(long)(k+1)*ldb];
    b1.x = B1[(long)k*ldb];  b1.y = B1[(long)(k+1)*ldb];
    if (k0 + 32 < g.K){
      __builtin_prefetch(B0 + (long)(k0+32)*ldb, 0, 1);  // global_prefetch_b8
      __builtin_prefetch(B1 + (long)(k0+32)*ldb, 0, 1);
    }
    acc0 = __builtin_amdgcn_wmma_f32_16x16x4_f32(false, a, false, b0, (short)0, acc0, false, false);
    acc1 = __builtin_amdgcn_wmma_f32_16x16x4_f32(false, a, false, b1, (short)0, acc1, false, false);
  }
  const int mb = blockIdx.x*16 + ((lane>>4)<<3);  // lanes 16-31 hold M=8..15
  if (c0){
    const float bs = g.bias ? g.bias[col0] : 0.0f;
#pragma unroll
    for (int r=0;r<8;r++){
      float vv = acc0[r] + bs;
      if (g.relu) vv = fmaxf(vv, 0.0f);
      g.C[(long)(mb+r)*g.ldc + col0] = vv;
    }
  }
  if (c1){
    const float bs = g.bias ? g.bias[col1] : 0.0f;
#pragma unroll
    for (int r=0;r<8;r++){
      float vv = acc1[r] + bs;
      if (g.relu) vv = fmaxf(vv, 0.0f);
      g.C[(long)(mb+r)*g.ldc + col1] = vv;
    }
  }
}

// Feat-gather GEMM for out1: A row = [h_f(512) | h_b(512) | attn1(128) | attn2(128)]
__global__ __launch_bounds__(32) void k_gemm_feat(GemmArgs g){
  const int lane = threadIdx.x;
  const int m16  = lane & 15;
  const int kb   = (lane >> 4) << 1;
  const int row  = blockIdx.x*16 + m16;
  const int col0 = blockIdx.y*32 + m16;
  const int col1 = col0 + 16;
  const bool c0 = col0 < g.Ncols, c1 = col1 < g.Ncols;
  const long ldb = g.Ncols;
  const float* B0 = g.B + (c0 ? col0 : (g.Ncols-1));
  const float* B1 = g.B + (c1 ? col1 : (g.Ncols-1));
  const float* segp[4]  = { g.f0 + (long)row*512, g.f1 + (long)row*512,
                            g.f2 + (long)row*128, g.f3 + (long)row*128 };
  const int    segl[4]  = { 512, 512, 128, 128 };
  v8f acc0 = {}, acc1 = {};
  int kg = 0;
#pragma unroll
  for (int s=0;s<4;s++){
    const float* Ap = segp[s];
    const int L = segl[s];
#pragma unroll 4
    for (int k0=0;k0<L;k0+=4){
      const int k = k0 + kb;
      v2f a = *(const v2f*)(Ap + k);
      const long krow = (long)(kg + k);
      v2f b0, b1;
      b0.x = B0[krow*ldb]; b0.y = B0[(krow+1)*ldb];
      b1.x = B1[krow*ldb]; b1.y = B1[(krow+1)*ldb];
      acc0 = __builtin_amdgcn_wmma_f32_16x16x4_f32(false, a, false, b0, (short)0, acc0, false, false);
      acc1 = __builtin_amdgcn_wmma_f32_16x16x4_f32(false, a, false, b1, (short)0, acc1, false, false);
    }
    kg += L;
  }
  const int mb = blockIdx.x*16 + ((lane>>4)<<3);
  if (c0){
    const float bs = g.bias ? g.bias[col0] : 0.0f;
#pragma unroll
    for (int r=0;r<8;r++){
      float vv = acc0[r] + bs;
      if (g.relu) vv = fmaxf(vv, 0.0f);
      g.C[(long)(mb+r)*g.ldc + col0] = vv;
    }
  }
  if (c1){
    const float bs = g.bias ? g.bias[col1] : 0.0f;
#pragma unroll
    for (int r=0;r<8;r++){
      float vv = acc1[r] + bs;
      if (g.relu) vv = fmaxf(vv, 0.0f);
      g.C[(long)(mb+r)*g.ldc + col1] = vv;
    }
  }
}

// ---------------------------------------------------------------------------
// Attention: one wave per query row, 512 keys spread 16/lane.
//   Q,K,V in (N, T, nh*hd); out_mode 0 -> O[(n*T+t)*stride + h*hd + d] (encoder)
//   out_mode 1 -> O[(t*N+n)*hd + d]  (ca_seq, nh==1)
// ---------------------------------------------------------------------------
struct AttnArgs {
  const float* Q; const float* K; const float* V;
  float* O;
  int nh, hd, stride, out_mode;
  float scale;
};

__global__ __launch_bounds__(256) void k_attn(AttnArgs a){
  __shared__ float qs[8][128];
  const int wid = threadIdx.x>>5, lane = threadIdx.x&31;
  const int n = blockIdx.x / a.nh, h = blockIdx.x % a.nh;
  const int t = blockIdx.y*8 + wid;
  const float* Qp = a.Q + ((long)n*TT + t)*a.stride + h*a.hd;
  const float* Kb = a.K + (long)n*TT*a.stride + h*a.hd;
  const float* Vb = a.V + (long)n*TT*a.stride + h*a.hd;
  for (int d=lane; d<a.hd; d+=32) qs[wid][d] = Qp[d]*a.scale;
  __syncthreads();
  float s[16];
#pragma unroll
  for (int kk=0;kk<16;kk++){
    const float* kp = Kb + (long)(lane + kk*32)*a.stride;
    float acc = 0.0f;
    for (int d=0; d<a.hd; d++) acc += qs[wid][d]*kp[d];
    s[kk]=acc;
  }
  float mx = -3.4e38f;
#pragma unroll
  for (int kk=0;kk<16;kk++) mx = fmaxf(mx, s[kk]);
  mx = wred_max(mx);
  float den = 0.0f;
#pragma unroll
  for (int kk=0;kk<16;kk++){ s[kk]=__expf(s[kk]-mx); den+=s[kk]; }
  den = wred_add(den);
  const float inv = 1.0f/den;
#pragma unroll
  for (int kk=0;kk<16;kk++) s[kk]*=inv;
  for (int d=0; d<a.hd; d++){
    float acc = 0.0f;
#pragma unroll
    for (int kk=0;kk<16;kk++) acc += s[kk]*Vb[(long)(lane+kk*32)*a.stride + d];
    acc = wred_add(acc);
    if (lane==0){
      long off = (a.out_mode==0) ? (((long)n*TT+t)*a.stride + h*a.hd + d)
                                 : (((long)t*NN+n)*a.hd + d);
      a.O[off]=acc;
    }
  }
}

// y = LN(A+R)*g + b  (wave per row, C<=128)
__global__ __launch_bounds__(128) void k_add_ln(const float* A, const float* R,
                                                const float* g, const float* b,
                                                float* Y, int C){
  const int row  = blockIdx.x*4 + (threadIdx.x>>5);
  const int lane = threadIdx.x&31;
  const float* ap = A + (long)row*C;
  const float* rp = R + (long)row*C;
  float v[4]; float mu=0.0f;
#pragma unroll
  for (int i=0;i<4;i++){ int c=lane+i*32; float x=(c<C)?(ap[c]+rp[c]):0.0f; v[i]=x; mu+=x; }
  mu = wred_add(mu)/(float)C;
  float var=0.0f;
#pragma unroll
  for (int i=0;i<4;i++){ int c=lane+i*32; if(c<C){ float d=v[i]-mu; var+=d*d; } }
  var = wred_add(var)/(float)C;
  const float inv = rsqrtf(var+1e-6f);
  float* yp = Y + (long)row*C;
#pragma unroll
  for (int i=0;i<4;i++){ int c=lane+i*32; if(c<C) yp[c]=(v[i]-mu)*inv*g[c]+b[c]; }
}

// (N,T,C) -> (T,N,C)
__global__ void k_nt2tn(const float* src, float* dst, int C, long total){
  long i = (long)blockIdx.x*256 + threadIdx.x;
  if (i>=total) return;
  int c = (int)(i % C); long r = i / C;
  int n = (int)(r & 31); int t = (int)(r >> 5);
  dst[i] = src[((long)n*TT + t)*C + c];
}

// xa[(n*T+t)*100+c] = x[(t*N+n)*1124 + 1024 + c]
__global__ void k_extract_xa(const float* x, float* xa){
  long i = (long)blockIdx.x*256 + threadIdx.x;
  if (i >= (long)ROWS*DA) return;
  int c = (int)(i % DA); long r = i / DA;
  int n = (int)(r>>9); int t = (int)(r&511);
  xa[i] = x[((long)t*NN+n)*(DR+DA) + DR + c];
}

__global__ void k_lengths(const float* umask, int* len){
  int n = threadIdx.x;
  if (n<NN){ float s=0.0f; for (int t=0;t<TT;t++) s += umask[(long)n*TT+t]; len[n]=(int)(s+0.5f); }
}

// dst[t,n,:] = src[len[n]-1-t, n, :] (or 0), arrays (T,N,C)
__global__ void k_reverse(const float* src, float* dst, const int* len, int C, long total){
  long i = (long)blockIdx.x*256 + threadIdx.x;
  if (i>=total) return;
  int c = (int)(i % C); long r = i / C;
  int n = (int)(r & 31); int t = (int)(r >> 5);
  int idx = len[n]-1-t;
  float v = 0.0f;
  if (idx>=0){ if (idx>TT-1) idx=TT-1; v = src[((long)idx*NN + n)*C + c]; }
  dst[i]=v;
}

// logits = out2(o1); log_softmax; write d_out[(n*T+t)*6+c]
__global__ __launch_bounds__(256) void k_head(const float* o1, const float* w2,
                                              const float* b2, float* out){
  int r = blockIdx.x*256 + threadIdx.x;
  if (r>=ROWS) return;
  float z[6];
#pragma unroll
  for (int c=0;c<6;c++) z[c]=b2[c];
  const float* op = o1 + (long)r*32;
  for (int j=0;j<32;j++){
    float x = op[j];
#pragma unroll
    for (int c=0;c<6;c++) z[c]+=x*w2[j*6+c];
  }
  float m=z[0];
#pragma unroll
  for (int c=1;c<6;c++) m=fmaxf(m,z[c]);
  float s=0.0f;
#pragma unroll
  for (int c=0;c<6;c++) s+=__expf(z[c]-m);
  float ls = __logf(s)+m;
  int t = r>>5, n = r&31;
#pragma unroll
  for (int c=0;c<6;c++) out[((long)n*TT+t)*6 + c] = z[c]-ls;
}

// ---------------------------------------------------------------------------
// MARN recurrent cell: persistent WG per (sample, direction). 512 serial steps.
// outer_ca uses the exact rank-1 factorization attn[i,j]=c_l[i]*wk[j]*<wq,c_a>.
// ---------------------------------------------------------------------------
struct CellW {
  const float *gih_w,*gih_b,*ghh_w,*ghh_b;
  const float *Wl,*Wlb,*Ul,*Ulb,*Vl,*Vlb,*Sl,*Slb;
  const float *Wa,*Wab,*Ua,*Uab,*Va,*Vab,*Sa,*Sab;
  const float *cawq,*cawk;
  const float *xl,*xa,*qm;   // (T,N,*) layouts
  float* hout;               // (T,N,512)
};

__global__ __launch_bounds__(256) void k_marn(CellW cf, CellW cb){
  CellW c = (blockIdx.y==0) ? cf : cb;
  const int n = blockIdx.x;
  const int tid = threadIdx.x;
  __shared__ float h_l[DHH],h_a[DHH],c_l[DHH],c_a[DHH],z_l[DHH],h_s[DHH];
  __shared__ float qmem[PP][DHH];
  __shared__ float Uin[DL+DA];
  __shared__ float gi[384], gh[384];
  __shared__ float sums[512];
  __shared__ float qm_s[PP];
  __shared__ float red[8];
  __shared__ float sS;
  __shared__ int   qidx;
  __shared__ float wq_s[DHH], wk_s[DHH];

  for (int i=tid;i<DHH;i+=256){
    h_l[i]=0;h_a[i]=0;c_l[i]=0;c_a[i]=0;z_l[i]=0;
    wq_s[i]=c.cawq[i]; wk_s[i]=c.cawk[i];
  }
  for (int i=tid;i<PP*DHH;i+=256) (&qmem[0][0])[i]=0.0f;
  __syncthreads();

  for (int t=0;t<TT;t++){
    const float* xlt = c.xl + ((long)t*NN+n)*DL;
    const float* xat = c.xa + ((long)t*NN+n)*DA;
    for (int i=tid;i<DL;i+=256) Uin[i]=xlt[i];
    for (int i=tid;i<DA;i+=256) Uin[DL+i]=xat[i];
    if (tid<PP) qm_s[tid]=c.qm[((long)t*NN+n)*PP+tid];
    __syncthreads();
    if (tid==0){
      int bi=0; float bv=qm_s[0];
      for (int p=1;p<PP;p++) if (qm_s[p]>bv){bv=qm_s[p];bi=p;}
      qidx=bi;
    }
    __syncthreads();
    const int qi=qidx;

    // GRU gates
    for (int j=tid;j<384;j+=256){
      float a=c.gih_b[j];
      for (int k2=0;k2<DL+DA;k2++) a+=Uin[k2]*c.gih_w[k2*384+j];
      float b=c.ghh_b[j];
      for (int k2=0;k2<DHH;k2++)   b+=qmem[qi][k2]*c.ghh_w[k2*384+j];
      gi[j]=a; gh[j]=b;
    }
    __syncthreads();
    for (int j=tid;j<DHH;j+=256){
      float r  = sigf(gi[j]+gh[j]);
      float z  = sigf(gi[128+j]+gh[128+j]);
      float nn2= tanhf(gi[256+j] + r*gh[256+j]);
      h_s[j] = (1.0f-z)*nn2 + z*qmem[qi][j];
    }
    __syncthreads();
    for (int i=tid;i<PP*DHH;i+=256){
      int p=i>>7, j=i&127;
      float m=qm_s[p];
      qmem[p][j] = qmem[p][j]*(1.0f-m) + h_s[j]*m;
    }
    __syncthreads();

    // LSTHM (language)
    for (int j=tid;j<512;j+=256){
      float a=c.Wlb[j]+c.Ulb[j]+c.Vlb[j]+c.Slb[j];
      for (int k2=0;k2<DL;k2++)  a+=Uin[k2]*c.Wl[k2*512+j];
      for (int k2=0;k2<DHH;k2++) a+=h_l[k2]*c.Ul[k2*512+j];
      for (int k2=0;k2<DHH;k2++) a+=z_l[k2]*c.Vl[k2*512+j];
      for (int k2=0;k2<DHH;k2++) a+=h_s[k2]*c.Sl[k2*512+j];
      sums[j]=a;
    }
    __syncthreads();
    for (int j=tid;j<DHH;j+=256){
      float f=sigf(sums[j]), ii=sigf(sums[128+j]), oo=sigf(sums[256+j]), ch=tanhf(sums[384+j]);
      float ct=f*c_l[j]+ii*ch;
      c_l[j]=ct; h_l[j]=tanhf(ct)*oo;
    }
    __syncthreads();

    // LSTHM (audio)
    for (int j=tid;j<512;j+=256){
      float a=c.Wab[j]+c.Uab[j]+c.Vab[j]+c.Sab[j];
      for (int k2=0;k2<DA;k2++)  a+=Uin[DL+k2]*c.Wa[k2*512+j];
      for (int k2=0;k2<DHH;k2++) a+=h_a[k2]*c.Ua[k2*512+j];
      for (int k2=0;k2<DHH;k2++) a+=z_l[k2]*c.Va[k2*512+j];
      for (int k2=0;k2<DHH;k2++) a+=h_s[k2]*c.Sa[k2*512+j];
      sums[j]=a;
    }
    __syncthreads();
    for (int j=tid;j<DHH;j+=256){
      float f=sigf(sums[j]), ii=sigf(sums[128+j]), oo=sigf(sums[256+j]), ch=tanhf(sums[384+j]);
      float ct=f*c_a[j]+ii*ch;
      c_a[j]=ct; h_a[j]=tanhf(ct)*oo;
    }
    __syncthreads();

    // outer_ca via rank-1 factorization
    float part=0.0f;
    for (int k2=tid;k2<DHH;k2+=256) part += wq_s[k2]*c_a[k2];
    part = wred_add(part);
    if ((tid&31)==0) red[tid>>5]=part;
    __syncthreads();
    if (tid==0){
      float s2=0.0f; for (int w=0;w<8;w++) s2+=red[w];
      sS = s2*0.088388347648318447f;   // 1/sqrt(128)
    }
    __syncthreads();
    for (int i=tid;i<DHH;i+=256){
      float ai=c_l[i]*sS;
      float mx=-3.4e38f;
      for (int j=0;j<DHH;j++) mx=fmaxf(mx, ai*wk_s[j]);
      float den=0.0f,num=0.0f;
      for (int j=0;j<DHH;j++){ float e=__expf(ai*wk_s[j]-mx); den+=e; num+=e*c_a[j]; }
      sums[i]=num/den;
    }
    __syncthreads();
    float* ho = c.hout + ((long)t*NN+n)*512;
    for (int j=tid;j<DHH;j+=256){
      float zn=sums[j];
      ho[j]=h_l[j]; ho[128+j]=h_a[j]; ho[256+j]=zn; ho[384+j]=h_s[j];
      z_l[j]=zn;
    }
    __syncthreads();
  }
}

// ---------------------------------------------------------------------------
// Host orchestration
// ---------------------------------------------------------------------------
static void gemm(hipStream_t st, const float* A, long lda, int amode, long acol,
                 const float* B, const float* bias, const float* alpha,
                 float* C, long ldc, int M, int Ncols, int K, int relu,
                 const float* f0=nullptr, const float* f1=nullptr,
                 const float* f2=nullptr, const float* f3=nullptr){
  GemmArgs g{A,lda,amode,acol,B,bias,alpha,C,ldc,M,Ncols,K,relu,f0,f1,f2,f3};
  dim3 grid((unsigned)(M/16), (unsigned)((Ncols+31)/32));
  if (amode==0)      k_gemm_wmma<0><<<grid,32,0,st>>>(g);
  else if (amode==1) k_gemm_wmma<1><<<grid,32,0,st>>>(g);
  else               k_gemm_feat<<<grid,32,0,st>>>(g);
}

extern "C" void kernel_launch(void* const* d_in, const int* in_sizes, int n_in,
                              void* d_out, int out_size, void* d_ws, size_t ws_size,
                              hipStream_t stream){
  (void)in_sizes; (void)n_in; (void)out_size; (void)ws_size;
  auto IN = [&](int i)->const float*{ return (const float*)d_in[i]; };

  float* W   = (float*)d_ws;
  float* xl  = W;                 // (N,T,100)
  float* xa  = W +  1638400;      // (N,T,100)
  float* qb  = W +  3276800;      // (N,T,320)
  float* kb  = W +  8519680;      // (N,T,320)
  float* vb  = W + 13762560;      // (N,T,320)
  float* ob  = W + 19005440;      // (N,T,320)
  float* t1  = W + 24248320;      // (N,T,100)
  float* mid = W + 25886720;      // (N,T,40)
  float* x1b = W + 26542080;      // (N,T,100)
  float* rxl = W + 28180480;      // (T,N,100)
  float* rxa = W + 29818880;      // (T,N,100)
  float* rqm = W + 31457280;      // (T,N,9)
  float* h_f = W + 31604736;      // (T,N,512)
  float* h_braw = W + 39993344;   // (T,N,512)
  float* h_b = W + 48381952;      // (T,N,512)
  float* at1 = W + 56770560;      // (T,N,128)
  float* at2 = W + 58867712;      // (T,N,128)
  float* o1  = W + 60964864;      // (T,N,32)
  int*   len = (int*)(W + 61489152);
  float* caQ = qb; float* caK = kb; float* caV = vb;     // reuse after encoders
  float* at1r = ob; float* at2r = ob + 2097152;          // (T,N,128) each

  float* outP  = (float*)d_out;
  float* xl_tn = outP + 98304;              // returned x_l (T,N,100)
  float* xa_tn = outP + 98304 + 1638400;    // returned x_a (T,N,100)

  const float* x_in  = IN(0);
  const float* qmask = IN(1);
  const float* umask = IN(2);

  k_lengths<<<1,32,0,stream>>>(umask, len);

  // linear_in: (T,N,1124)[:,:, :1024] perm -> (N,T,100)
  gemm(stream, x_in, DR+DA, 1, 0, IN(92), IN(91), nullptr, xl, DL, ROWS, DL, DR, 0);
  k_extract_xa<<<(ROWS*DA+255)/256,256,0,stream>>>(x_in, xa);

  // encoder passes (enc_l twice on xl, enc_a twice on xa)
  auto encp = [&](float* xbuf, int be){
    const float *fc_b=IN(be+0), *fc_w=IN(be+1);
    const float *ln1_b=IN(be+2), *ln1_g=IN(be+3);
    const float *ln2_b=IN(be+4), *ln2_g=IN(be+5);
    const float *w1_b=IN(be+6), *w1_w=IN(be+7);
    const float *w2_b=IN(be+8), *w2_w=IN(be+9);
    const float *wk_b=IN(be+10), *wk_w=IN(be+11);
    const float *wq_b=IN(be+12), *wq_w=IN(be+13);
    const float *wv_b=IN(be+14), *wv_w=IN(be+15);
    gemm(stream, xbuf, DL, 0, 0, wq_w, wq_b, nullptr, qb, 320, ROWS, 320, DL, 0);
    gemm(stream, xbuf, DL, 0, 0, wk_w, wk_b, nullptr, kb, 320, ROWS, 320, DL, 0);
    gemm(stream, xbuf, DL, 0, 0, wv_w, wv_b, nullptr, vb, 320, ROWS, 320, DL, 0);
    AttnArgs aa{qb,kb,vb,ob,NHD,DKH,320,0, 0.15811388300841898f}; // 1/sqrt(40)
    k_attn<<<dim3(NN*NHD,64),256,0,stream>>>(aa);
    gemm(stream, ob, 320, 0, 0, fc_w, fc_b, nullptr, t1, DL, ROWS, DL, 320, 0);
    k_add_ln<<<ROWS/4,128,0,stream>>>(t1, xbuf, ln1_g, ln1_b, x1b, DL);
    gemm(stream, x1b, DL, 0, 0, w1_w, w1_b, nullptr, mid, 40, ROWS, 40, DL, 1);
    gemm(stream, mid, 40, 0, 0, w2_w, w2_b, nullptr, t1, DL, ROWS, DL, 40, 0);
    k_add_ln<<<ROWS/4,128,0,stream>>>(t1, x1b, ln2_g, ln2_b, xbuf, DL);
  };
  encp(xl, 75); encp(xa, 59); encp(xl, 75); encp(xa, 59);

  // (N,T,100) -> (T,N,100), written directly into d_out (also used downstream)
  k_nt2tn<<<(ROWS*DL+255)/256,256,0,stream>>>(xl, xl_tn, DL, (long)ROWS*DL);
  k_nt2tn<<<(ROWS*DL+255)/256,256,0,stream>>>(xa, xa_tn, DL, (long)ROWS*DL);

  // reversed inputs for backward cell
  k_reverse<<<(ROWS*DL+255)/256,256,0,stream>>>(xl_tn, rxl, len, DL, (long)ROWS*DL);
  k_reverse<<<(ROWS*DL+255)/256,256,0,stream>>>(xa_tn, rxa, len, DL, (long)ROWS*DL);
  k_reverse<<<(ROWS*PP+255)/256,256,0,stream>>>(qmask, rqm, len, PP, (long)ROWS*PP);

  auto cellw = [&](int b, const float* xlp, const float* xap, const float* qmp, float* hout){
    CellW c;
    c.cawk=IN(b+0);  c.cawq=IN(b+1);
    c.ghh_b=IN(b+2); c.ghh_w=IN(b+3);
    c.gih_b=IN(b+4); c.gih_w=IN(b+5);
    c.Sab=IN(b+6);  c.Sa=IN(b+7);  c.Uab=IN(b+8);  c.Ua=IN(b+9);
    c.Vab=IN(b+10); c.Va=IN(b+11); c.Wab=IN(b+12); c.Wa=IN(b+13);
    c.Slb=IN(b+14); c.Sl=IN(b+15); c.Ulb=IN(b+16); c.Ul=IN(b+17);
    c.Vlb=IN(b+18); c.Vl=IN(b+19); c.Wlb=IN(b+20); c.Wl=IN(b+21);
    c.xl=xlp; c.xa=xap; c.qm=qmp; c.hout=hout;
    return c;
  };
  CellW cfw = cellw(37, xl_tn, xa_tn, qmask, h_f);      // cell_f
  CellW cbw = cellw(15, rxl,   rxa,   rqm,   h_braw);   // cell_b
  k_marn<<<dim3(NN,2),256,0,stream>>>(cfw, cbw);
  k_reverse<<<(ROWS*512+255)/256,256,0,stream>>>(h_braw, h_b, len, 512, (long)ROWS*512);

  const float* pw  = IN(100); const float* pv = IN(97);
  const float* pv1 = IN(98);  const float* pv2 = IN(99);
  const float caS = 0.088388347648318447f;  // 1/sqrt(128)

  // ca2_l2a: Q=w*x_l, K/V=v*x_a
  gemm(stream, xl_tn, DL, 1, 0, IN(7), nullptr, pw, caQ, 128, ROWS, 128, DL, 0);
  gemm(stream, xa_tn, DL, 1, 0, IN(6), nullptr, pv, caK, 128, ROWS, 128, DL, 0);
  gemm(stream, xa_tn, DL, 1, 0, IN(8), nullptr, pv, caV, 128, ROWS, 128, DL, 0);
  { AttnArgs aa{caQ,caK,caV,at1r,1,128,128,1,caS}; k_attn<<<dim3(NN,64),256,0,stream>>>(aa); }
  // ca2_a2l: Q=v*x_a, K/V=w*x_l
  gemm(stream, xa_tn, DL, 1, 0, IN(4), nullptr, pv, caQ, 128, ROWS, 128, DL, 0);
  gemm(stream, xl_tn, DL, 1, 0, IN(3), nullptr, pw, caK, 128, ROWS, 128, DL, 0);
  gemm(stream, xl_tn, DL, 1, 0, IN(5), nullptr, pw, caV, 128, ROWS, 128, DL, 0);
  { AttnArgs aa{caQ,caK,caV,at2r,1,128,128,1,caS}; k_attn<<<dim3(NN,64),256,0,stream>>>(aa); }
  // ca3_l2a: Q=v*x_a, K/V=v1*attn1
  gemm(stream, xa_tn, DL, 1, 0, IN(13), nullptr, pv,  caQ, 128, ROWS, 128, DL, 0);
  gemm(stream, at1r, 128, 1, 0, IN(12), nullptr, pv1, caK, 128, ROWS, 128, 128, 0);
  gemm(stream, at1r, 128, 1, 0, IN(14), nullptr, pv1, caV, 128, ROWS, 128, 128, 0);
  { AttnArgs aa{caQ,caK,caV,at1,1,128,128,1,caS}; k_attn<<<dim3(NN,64),256,0,stream>>>(aa); }
  // ca3_a2l: Q=w*x_l, K/V=v2*attn2
  gemm(stream, xl_tn, DL, 1, 0, IN(10), nullptr, pw,  caQ, 128, ROWS, 128, DL, 0);
  gemm(stream, at2r, 128, 1, 0, IN(9),  nullptr, pv2, caK, 128, ROWS, 128, 128, 0);
  gemm(stream, at2r, 128, 1, 0, IN(11), nullptr, pv2, caV, 128, ROWS, 128, 128, 0);
  { AttnArgs aa{caQ,caK,caV,at2,1,128,128,1,caS}; k_attn<<<dim3(NN,64),256,0,stream>>>(aa); }

  // out1 over gathered 1280-wide features, fused ReLU
  gemm(stream, nullptr, 0, 2, 0, IN(94), IN(93), nullptr, o1, 32, ROWS, 32, 1280, 1,
       h_f, h_b, at1, at2);
  // out2 + log_softmax + (T,N)->(N,T) reindex
  k_head<<<(ROWS+255)/256,256,0,stream>>>(o1, IN(96), IN(95), outP);
}